// bin_loss_73358041415724
// MI455X (gfx1250) — compile-verified
//
#include <hip/hip_runtime.h>
#include <hip/hip_bf16.h>

typedef float v2f __attribute__((ext_vector_type(2)));
typedef float v4f __attribute__((ext_vector_type(4)));
typedef float v8f __attribute__((ext_vector_type(8)));

#ifndef LOSS_NB
#define LOSS_NB 2048
#endif
#define LOSS_TPB 256

// Kernel 1: streaming partial reduction of sum(k^2 + (k-1)^2).
// - grid-stride loop over float4 (global_load_b128, non-temporal)
// - per-thread dual fp32 accumulators (ILP)
// - wave-level reduction via V_WMMA_F32_16X16X4_F32 with A = ones:
//     D[m][n] = sum_k B[k][n]  =>  sum over all 256 D elements = 16 * sum(B)
//   (layout-robust: independent of exact lane<->(k,n) mapping)
// - LDS tree across the 8 waves of the block; one fp32 partial per block.
__global__ void __launch_bounds__(LOSS_TPB)
bin_loss_partial_kernel(const float* __restrict__ k,
                        float* __restrict__ partial,
                        long long n4, long long n) {
    const int tid = threadIdx.x;
    const long long gid    = (long long)blockIdx.x * blockDim.x + tid;
    const long long stride = (long long)gridDim.x * blockDim.x;

    const v4f* __restrict__ k4 = (const v4f*)k;
    float acc0 = 0.0f, acc1 = 0.0f;

    for (long long i = gid; i < n4; i += stride) {
        v4f v = __builtin_nontemporal_load(&k4[i]);   // global_load_b128, TH=NT
        float dx = v.x - 1.0f;
        float dy = v.y - 1.0f;
        float dz = v.z - 1.0f;
        float dw = v.w - 1.0f;
        acc0 = __builtin_fmaf(v.x, v.x, acc0);
        acc1 = __builtin_fmaf(v.y, v.y, acc1);
        acc0 = __builtin_fmaf(v.z, v.z, acc0);
        acc1 = __builtin_fmaf(v.w, v.w, acc1);
        acc0 = __builtin_fmaf(dx, dx, acc0);
        acc1 = __builtin_fmaf(dy, dy, acc1);
        acc0 = __builtin_fmaf(dz, dz, acc0);
        acc1 = __builtin_fmaf(dw, dw, acc1);
    }

    // Scalar tail (n % 4 != 0) handled by global thread 0; EXEC reconverges
    // to all-ones before the WMMA below (structured if).
    if (gid == 0) {
        for (long long i = n4 * 4; i < n; ++i) {
            float x = k[i];
            float d = x - 1.0f;
            acc0 = __builtin_fmaf(x, x, acc0);
            acc0 = __builtin_fmaf(d, d, acc0);
        }
    }

    // Wave-level 64->16 reduction on the matrix pipe (fp32 end to end).
    v2f a; a.x = 1.0f; a.y = 1.0f;     // A = ones(16x4)
    v2f b; b.x = acc0; b.y = acc1;     // B = all 64 per-wave accumulators
    v8f c = {};
    v8f d = __builtin_amdgcn_wmma_f32_16x16x4_f32(
        /*neg_a=*/false, a, /*neg_b=*/false, b,
        /*c_mod=*/(short)0, c, /*reuse_a=*/false, /*reuse_b=*/false);

    // Per-lane sum of its 8 D registers; block sum of these = 16 * block_total.
    float ds = ((d[0] + d[1]) + (d[2] + d[3])) + ((d[4] + d[5]) + (d[6] + d[7]));

    __shared__ float sm[LOSS_TPB];
    sm[tid] = ds;
    __syncthreads();
    for (int s = LOSS_TPB / 2; s > 0; s >>= 1) {
        if (tid < s) sm[tid] += sm[tid + s];
        __syncthreads();
    }
    if (tid == 0) partial[blockIdx.x] = sm[0] * (1.0f / 16.0f);  // exact /16
}

// Kernel 2: reduce the per-block partials in double, write mean as fp32.
__global__ void __launch_bounds__(LOSS_TPB)
bin_loss_finalize_kernel(const float* __restrict__ partial, int nb,
                         float* __restrict__ out, double inv_n) {
    __shared__ double sm[LOSS_TPB];
    const int tid = threadIdx.x;
    double acc = 0.0;
    for (int i = tid; i < nb; i += LOSS_TPB) acc += (double)partial[i];
    sm[tid] = acc;
    __syncthreads();
    for (int s = LOSS_TPB / 2; s > 0; s >>= 1) {
        if (tid < s) sm[tid] += sm[tid + s];
        __syncthreads();
    }
    if (tid == 0) out[0] = (float)(sm[0] * inv_n);
}

extern "C" void kernel_launch(void* const* d_in, const int* in_sizes, int n_in,
                              void* d_out, int out_size, void* d_ws, size_t ws_size,
                              hipStream_t stream) {
    (void)n_in; (void)out_size;
    const float* k = (const float*)d_in[0];
    const long long n  = (long long)in_sizes[0];
    const long long n4 = n / 4;

    int nb = LOSS_NB;
    const long long max_nb = (long long)(ws_size / sizeof(float));
    if (max_nb < nb) nb = (int)(max_nb > 0 ? max_nb : 1);

    float* partial = (float*)d_ws;

    bin_loss_partial_kernel<<<nb, LOSS_TPB, 0, stream>>>(k, partial, n4, n);
    bin_loss_finalize_kernel<<<1, LOSS_TPB, 0, stream>>>(
        partial, nb, (float*)d_out, 1.0 / (double)n);
}